// Dynamic_ConvLayer2_59820304498803
// MI455X (gfx1250) — compile-verified
//
#include <hip/hip_runtime.h>

#define IN_CH   64
#define OUT_CH  64
#define KS      3
#define GROUPS  8
#define FPG     8          // channels per group (== out-ch per group)
#define HDIM    256
#define WDIM    256
#define BATCH   8
#define KVEC    (OUT_CH * KS * KS)   // 576
#define RDIM    32

typedef __attribute__((ext_vector_type(2))) float v2f;
typedef __attribute__((ext_vector_type(8))) float v8f;

// ---------------------------------------------------------------------------
// Kernel 1: kern[b][o*9+t] = leaky_relu(rep[b,:] . W[o*9+t,:], 0.1)
// GEMM M=8(pad to 16) x N=576 x K=32 via V_WMMA_F32_16X16X4_F32 (8 K-steps).
// A 16x4 f32 layout: lanes 0-15 hold M=lane, VGPR0={K=k0 | K=k0+2 for lanes
// 16-31}, VGPR1={K=k0+1 | K=k0+3}.  B 4x16 mirrors with N on lanes.
// D 16x16 f32: VGPR j -> M=j (lanes 0-15), M=j+8 (lanes 16-31, padding here).
// All fragments preloaded (branch-free, mask-multiplied zero padding) so the
// 8 global_load_b64 issue together and the 8 WMMAs run back-to-back.
// ---------------------------------------------------------------------------
__global__ __launch_bounds__(32) void gen_filters(const float* __restrict__ rep,
                                                  const float* __restrict__ Wm,
                                                  float* __restrict__ kern) {
  const int lane  = threadIdx.x;        // 0..31
  const int n0    = blockIdx.x * 16;    // 36 N-tiles of 16
  const int mn    = lane & 15;          // M row (A) / N col (B)
  const int khalf = (lane >> 4) << 1;   // 0 (lanes 0-15) or 2 (lanes 16-31)

#if __has_builtin(__builtin_amdgcn_wmma_f32_16x16x4_f32)
  const float amask = (mn < BATCH) ? 1.f : 0.f;   // zero-pad M rows 8..15
  const int   arow  = (mn < BATCH) ? mn : 0;      // clamped, always-valid row

  v2f A[8], B[8];
  #pragma unroll
  for (int s = 0; s < 8; ++s) {
    const int k0 = s * 4;
    A[s].x = amask * rep[arow * RDIM + k0 + khalf];
    A[s].y = amask * rep[arow * RDIM + k0 + khalf + 1];
    B[s].x = Wm[(n0 + mn) * RDIM + k0 + khalf];
    B[s].y = Wm[(n0 + mn) * RDIM + k0 + khalf + 1];
  }
  v8f c = {};
  #pragma unroll
  for (int s = 0; s < 8; ++s)
    c = __builtin_amdgcn_wmma_f32_16x16x4_f32(false, A[s], false, B[s],
                                              (short)0, c, false, false);
  if (lane < 16) {
    #pragma unroll
    for (int j = 0; j < BATCH; ++j) {   // valid M rows 0..7 live in lanes 0-15
      float v = c[j];
      kern[j * KVEC + n0 + lane] = (v >= 0.f) ? v : 0.1f * v;
    }
  }
#else
  if (lane < 16) {
    for (int j = 0; j < BATCH; ++j) {
      float acc = 0.f;
      for (int k = 0; k < RDIM; ++k)
        acc += rep[j * RDIM + k] * Wm[(n0 + lane) * RDIM + k];
      kern[j * KVEC + n0 + lane] = (acc >= 0.f) ? acc : 0.1f * acc;
    }
  }
#endif
}

// ---------------------------------------------------------------------------
// Kernel 2: fused reflection-pad + group-channel-sum + 9-tap dynamic stencil.
// out[b,o,h,w] = sum_{kh,kw} k[b,o,kh,kw] * (sum_{ic in group} xp[b,ic,h+kh,w+kw])
// Block = (batch b, group g, 16x64 output tile).  Each x element touches HBM
// exactly once; group sum accumulated in registers then staged through LDS.
// ~270 MB total traffic -> ~12 us floor at 23.3 TB/s; FLOPs (0.6 G) are free.
// ---------------------------------------------------------------------------
#define TH 16
#define TW 64
#define HALO_H (TH + 2)                // 18
#define HALO_W (TW + 2)                // 66
#define HALO_N (HALO_H * HALO_W)       // 1188
#define NLOAD  ((HALO_N + 255) / 256)  // 5

__global__ __launch_bounds__(256) void dyn_conv(const float* __restrict__ x,
                                                const float* __restrict__ kern,
                                                float* __restrict__ out) {
  __shared__ float s_tile[HALO_N];
  __shared__ float s_kern[FPG * 9];

  const int tid = threadIdx.x;
  const int w0  = blockIdx.x * TW;
  const int h0  = blockIdx.y * TH;
  const int bz  = blockIdx.z;
  const int b   = bz >> 3;
  const int g   = bz & 7;

  if (tid < FPG * 9)
    s_kern[tid] = kern[b * KVEC + (g * FPG) * 9 + tid];

  // Precompute reflect-padded global offsets for this thread's halo slots.
  int off[NLOAD];
  #pragma unroll
  for (int i = 0; i < NLOAD; ++i) {
    int idx = tid + i * 256;
    int r   = idx / HALO_W;
    int cc  = idx - r * HALO_W;
    int gh  = h0 - 1 + r;                 // reflect pad=1: -1 -> 1, 256 -> 254
    gh = (gh < 0) ? 1 : ((gh > HDIM - 1) ? HDIM - 2 : gh);
    int gw  = w0 - 1 + cc;
    gw = (gw < 0) ? 1 : ((gw > WDIM - 1) ? WDIM - 2 : gw);
    off[i] = gh * WDIM + gw;
  }

  float acc[NLOAD];
  #pragma unroll
  for (int i = 0; i < NLOAD; ++i) acc[i] = 0.f;

  const float* xg = x + (size_t)(b * IN_CH + g * FPG) * (HDIM * WDIM);
  #pragma unroll
  for (int ic = 0; ic < FPG; ++ic) {
    const float* xc = xg + (size_t)ic * (HDIM * WDIM);
    #pragma unroll
    for (int i = 0; i < NLOAD; ++i) {
      int idx = tid + i * 256;
      if (idx < HALO_N) acc[i] += xc[off[i]];
    }
  }
  #pragma unroll
  for (int i = 0; i < NLOAD; ++i) {
    int idx = tid + i * 256;
    if (idx < HALO_N) s_tile[idx] = acc[i];
  }
  __syncthreads();

  const int wo   = tid >> 5;   // wave -> out channel within group (wave32)
  const int lane = tid & 31;

  float kf[9];
  #pragma unroll
  for (int t = 0; t < 9; ++t) kf[t] = s_kern[wo * 9 + t];

  float* outc = out + ((size_t)(b * OUT_CH + g * FPG + wo) * HDIM + h0) * WDIM + w0;

  #pragma unroll
  for (int half = 0; half < 2; ++half) {
    const int c = lane + half * 32;      // output column within tile, 0..63
    // Vertical sliding window: 3 LDS reads per output, 6 values reused.
    float r0a = s_tile[0 * HALO_W + c];
    float r0b = s_tile[0 * HALO_W + c + 1];
    float r0c = s_tile[0 * HALO_W + c + 2];
    float r1a = s_tile[1 * HALO_W + c];
    float r1b = s_tile[1 * HALO_W + c + 1];
    float r1c = s_tile[1 * HALO_W + c + 2];
    #pragma unroll
    for (int h = 0; h < TH; ++h) {
      float r2a = s_tile[(h + 2) * HALO_W + c];
      float r2b = s_tile[(h + 2) * HALO_W + c + 1];
      float r2c = s_tile[(h + 2) * HALO_W + c + 2];
      float v = kf[0] * r0a + kf[1] * r0b + kf[2] * r0c
              + kf[3] * r1a + kf[4] * r1b + kf[5] * r1c
              + kf[6] * r2a + kf[7] * r2b + kf[8] * r2c;
      outc[h * WDIM + c] = v;
      r0a = r1a; r0b = r1b; r0c = r1c;
      r1a = r2a; r1b = r2b; r1c = r2c;
    }
  }
}

extern "C" void kernel_launch(void* const* d_in, const int* in_sizes, int n_in,
                              void* d_out, int out_size, void* d_ws, size_t ws_size,
                              hipStream_t stream) {
  const float* x    = (const float*)d_in[0];  // [8,64,256,256]
  const float* rep  = (const float*)d_in[1];  // [8,32]
  const float* Wm   = (const float*)d_in[2];  // [576,32]
  float*       out  = (float*)d_out;          // [8,64,256,256]
  float*       kern = (float*)d_ws;           // [8,576] = 18432 bytes scratch

  gen_filters<<<dim3(KVEC / 16), 32, 0, stream>>>(rep, Wm, kern);
  dyn_conv<<<dim3(WDIM / TW, HDIM / TH, BATCH * GROUPS), 256, 0, stream>>>(x, kern, out);
}